// SubGNN_27986006901350
// MI455X (gfx1250) — compile-verified
//
#include <hip/hip_runtime.h>
#include <hip/hip_bf16.h>

typedef __attribute__((ext_vector_type(2))) float v2f;
typedef __attribute__((ext_vector_type(8))) float v8f;

#define BB 2
#define NN 16
#define MM 8
#define DD2 4
#define HH 32

// Workspace layout (floats)
static constexpr size_t OFF_F     = 0;                        // 512 x 132 shared enc features
static constexpr size_t OFF_ZSUM  = OFF_F + 512*132;          // 2 x 24
static constexpr size_t OFF_IP    = OFF_ZSUM + 48;            // 2 x 16 x 8
static constexpr size_t OFF_CENTH = OFF_IP + 256;             // 2 x 16 x 32
static constexpr size_t OFF_W1E1  = OFF_CENTH + 1024;         // 132 x 128 (padded enc1 W0)
static constexpr size_t OFF_W1E2  = OFF_W1E1 + 132*128;       // 132 x 128 (padded enc2 W0)
static constexpr size_t OFF_W1D   = OFF_W1E2 + 132*128;       // 92 x 128  (padded dec W0)
static constexpr size_t OFF_W5E1  = OFF_W1D + 92*128;         // 128 w5s + b5s at [128]
static constexpr size_t OFF_W5E2  = OFF_W5E1 + 132;
static constexpr size_t OFF_W5D   = OFF_W5E2 + 132;
static constexpr size_t OFF_YA    = OFF_W5D + 132;            // 2 x 512 x 128 ping
static constexpr size_t OFF_YB    = OFF_YA + 2*512*128;       // 2 x 512 x 128 pong
static constexpr size_t OFF_S     = OFF_YB + 2*512*128;       // 2 x 512
static constexpr size_t OFF_SROW  = OFF_S + 1024;             // 2 x 2 x 16
static constexpr size_t OFF_COVAR = OFF_SROW + 64;            // 2 x 16 x 32
static constexpr size_t OFF_CSUM  = OFF_COVAR + 1024;         // 2 x 32
static constexpr size_t OFF_FD    = OFF_CSUM + 64;            // 32 x 92 dec features
static constexpr size_t OFF_YDA   = OFF_FD + 32*92;           // 32 x 128
static constexpr size_t OFF_YDB   = OFF_YDA + 32*128;         // 32 x 128

// ---------------------------------------------------------------------------
// Prep: mask averages, centering, ip, shared enc feature matrix F (512x132),
// Zsum, zero-padded first-layer weights, collapsed layer-5 weights.
// ---------------------------------------------------------------------------
__global__ void __launch_bounds__(256)
prep_kernel(const float* __restrict__ z, const float* __restrict__ h,
            const float* __restrict__ masks,
            const float* __restrict__ W0e1, const float* __restrict__ W0e2,
            const float* __restrict__ W0d,
            const float* __restrict__ W4e1, const float* __restrict__ b4e1,
            const float* __restrict__ W4e2, const float* __restrict__ b4e2,
            const float* __restrict__ W4d,  const float* __restrict__ b4d,
            float* __restrict__ ws) {
  __shared__ float s_mz[BB][MM][DD2];
  __shared__ float s_mh[BB][MM][HH];
  __shared__ float s_centz[BB][NN][DD2];
  __shared__ float s_centh[BB][NN][HH];
  __shared__ float s_ip[BB][NN][8];
  const int tid = threadIdx.x;

  for (int idx = tid; idx < BB*MM*DD2; idx += blockDim.x) {
    int b = idx / (MM*DD2), m = (idx / DD2) % MM, d = idx % DD2;
    float num = 0.f, den = 0.f;
    for (int n = 0; n < NN; ++n) {
      float w = masks[(b*NN+n)*MM+m];
      num += w * z[(b*NN+n)*DD2+d]; den += w;
    }
    s_mz[b][m][d] = num / den;
  }
  for (int idx = tid; idx < BB*MM*HH; idx += blockDim.x) {
    int b = idx / (MM*HH), m = (idx / HH) % MM, d = idx % HH;
    float num = 0.f, den = 0.f;
    for (int n = 0; n < NN; ++n) {
      float w = masks[(b*NN+n)*MM+m];
      num += w * h[(b*NN+n)*HH+d]; den += w;
    }
    s_mh[b][m][d] = num / den;
  }
  __syncthreads();
  for (int idx = tid; idx < BB*NN*DD2; idx += blockDim.x) {
    int b = idx / (NN*DD2), n = (idx / DD2) % NN, d = idx % DD2;
    float acc = 0.f;
    for (int m = 0; m < MM; ++m) acc += masks[(b*NN+n)*MM+m] * s_mz[b][m][d];
    s_centz[b][n][d] = acc;
  }
  for (int idx = tid; idx < BB*NN*HH; idx += blockDim.x) {
    int b = idx / (NN*HH), n = (idx / HH) % NN, d = idx % HH;
    float acc = 0.f;
    for (int m = 0; m < MM; ++m) acc += masks[(b*NN+n)*MM+m] * s_mh[b][m][d];
    s_centh[b][n][d] = acc;
    ws[OFF_CENTH + idx] = acc;
  }
  __syncthreads();
  // ip = [z - cent_z (4), z[...,2:] (2), cent_z[...,2:] (2)]
  for (int idx = tid; idx < BB*NN; idx += blockDim.x) {
    int b = idx / NN, n = idx % NN;
    const float* zp = &z[(b*NN+n)*DD2];
    float ipv[8];
    ipv[0] = zp[0]-s_centz[b][n][0]; ipv[1] = zp[1]-s_centz[b][n][1];
    ipv[2] = zp[2]-s_centz[b][n][2]; ipv[3] = zp[3]-s_centz[b][n][3];
    ipv[4] = zp[2]; ipv[5] = zp[3];
    ipv[6] = s_centz[b][n][2]; ipv[7] = s_centz[b][n][3];
    for (int c = 0; c < 8; ++c) { s_ip[b][n][c] = ipv[c]; ws[OFF_IP + (b*NN+n)*8 + c] = ipv[c]; }
  }
  __syncthreads();
  // F rows: [gram_diag, h[m], h[n], cent_h[m], cent_h[n], 0,0,0]
  for (int p = tid; p < BB*NN*NN; p += blockDim.x) {
    int b = p / (NN*NN), n = (p / NN) % NN, m = p % NN;
    float cz[24];
    for (int c = 0; c < 8; ++c) cz[c]   = s_ip[b][n][c];
    for (int c = 0; c < 8; ++c) cz[8+c] = s_ip[b][m][c];
    for (int c = 0; c < 4; ++c) cz[16+c] = z[(b*NN+m)*DD2+c] - z[(b*NN+n)*DD2+c];
    cz[20] = z[(b*NN+m)*DD2+2]; cz[21] = z[(b*NN+m)*DD2+3];
    cz[22] = z[(b*NN+n)*DD2+2]; cz[23] = z[(b*NN+n)*DD2+3];
    float g = 1.0f;                       // + ||[0,0,-1]||^2
    for (int c = 0; c < 24; ++c) g += cz[c]*cz[c];
    float* row = &ws[OFF_F + (size_t)p * 132];
    row[0] = g;
    for (int c = 0; c < HH; ++c) row[1+c]      = h[(b*NN+m)*HH+c];
    for (int c = 0; c < HH; ++c) row[1+HH+c]   = h[(b*NN+n)*HH+c];
    for (int c = 0; c < HH; ++c) row[1+2*HH+c] = s_centh[b][m][c];
    for (int c = 0; c < HH; ++c) row[1+3*HH+c] = s_centh[b][n][c];
    row[129] = 0.f; row[130] = 0.f; row[131] = 0.f;
  }
  // Zsum[b,d] = sum over all 256 pairs of cat_z component d
  for (int idx = tid; idx < BB*24; idx += blockDim.x) {
    int b = idx / 24, d = idx % 24;
    float acc = 0.f;
    for (int n = 0; n < NN; ++n)
      for (int m = 0; m < NN; ++m) {
        float val;
        if (d < 8)       val = s_ip[b][n][d];
        else if (d < 16) val = s_ip[b][m][d-8];
        else if (d < 20) val = z[(b*NN+m)*DD2 + (d-16)] - z[(b*NN+n)*DD2 + (d-16)];
        else if (d < 22) val = z[(b*NN+m)*DD2 + (d-18)];
        else             val = z[(b*NN+n)*DD2 + (d-20)];
        acc += val;
      }
    ws[OFF_ZSUM + idx] = acc;
  }
  // Zero-padded first-layer weights (K: 129->132, 89->92)
  for (int idx = tid; idx < 132*128; idx += blockDim.x) {
    int k = idx / 128;
    ws[OFF_W1E1 + idx] = (k < 129) ? W0e1[idx] : 0.f;
    ws[OFF_W1E2 + idx] = (k < 129) ? W0e2[idx] : 0.f;
  }
  for (int idx = tid; idx < 92*128; idx += blockDim.x) {
    int k = idx / 128;
    ws[OFF_W1D + idx] = (k < 89) ? W0d[idx] : 0.f;
  }
  // Collapsed final layer: w5s[k] = sum_o W4[k,o]; b5s = sum_o b4[o]
  for (int k = tid; k < 128; k += blockDim.x) {
    float a1 = 0.f, a2 = 0.f, a3 = 0.f;
    for (int o = 0; o < 128; ++o) {
      a1 += W4e1[k*128+o]; a2 += W4e2[k*128+o]; a3 += W4d[k*128+o];
    }
    ws[OFF_W5E1+k] = a1; ws[OFF_W5E2+k] = a2; ws[OFF_W5D+k] = a3;
  }
  if (tid == 0) {
    float a1 = 0.f, a2 = 0.f, a3 = 0.f;
    for (int o = 0; o < 128; ++o) { a1 += b4e1[o]; a2 += b4e2[o]; a3 += b4d[o]; }
    ws[OFF_W5E1+128] = a1; ws[OFF_W5E2+128] = a2; ws[OFF_W5D+128] = a3;
  }
}

// ---------------------------------------------------------------------------
// fp32 WMMA GEMM: Y = act(X @ W + b), one wave computes one 16x16 tile.
// blockIdx.z selects one of two parameter sets (enc1/enc2).
// A fragment (16x4): lane m=lane&15; VGPR0/1 hold K={k,k+1} (lanes<16) or
// K={k+2,k+3} (lanes>=16). B/C/D row-striped across lanes.
// ---------------------------------------------------------------------------
__global__ void __launch_bounds__(32)
gemm_relu_wmma(const float* __restrict__ X0, const float* __restrict__ X1,
               const float* __restrict__ W0p, const float* __restrict__ W1p,
               const float* __restrict__ b0p, const float* __restrict__ b1p,
               float* __restrict__ Y0, float* __restrict__ Y1,
               int K, int ldx, int relu) {
  const int s = blockIdx.z;
  const float* X    = s ? X1  : X0;
  const float* W    = s ? W1p : W0p;
  const float* bias = s ? b1p : b0p;
  float*       Y    = s ? Y1  : Y0;
  const int tileM = blockIdx.x * 16;
  const int tileN = blockIdx.y * 16;
  const int lane  = threadIdx.x;
  const int half  = lane >> 4;
  const int l16   = lane & 15;
  v8f acc = {};
  for (int k = 0; k < K; k += 4) {
    const int kb = k + 2*half;
    v2f a, bf;
    const float* xr = &X[(size_t)(tileM + l16) * ldx + kb];
    a.x = xr[0];
    a.y = xr[1];
    bf.x = W[(size_t)kb       * 128 + tileN + l16];
    bf.y = W[(size_t)(kb + 1) * 128 + tileN + l16];
    acc = __builtin_amdgcn_wmma_f32_16x16x4_f32(
        /*neg_a=*/false, a, /*neg_b=*/false, bf,
        /*c_mod=*/(short)0, acc, /*reuse_a=*/false, /*reuse_b=*/false);
  }
  const int col = tileN + l16;
  const float bv = bias[col];
#pragma unroll
  for (int i = 0; i < 8; ++i) {
    const int row = tileM + i + 8*half;
    float y = acc[i] + bv;
    if (relu) y = fmaxf(y, 0.f);
    Y[(size_t)row * 128 + col] = y;
  }
}

// ---------------------------------------------------------------------------
// Combine: collapsed layer-5 dot, m-sums, rank-1 e1/e2, covar, Csum, dec feats
// ---------------------------------------------------------------------------
__global__ void __launch_bounds__(256)
combine_kernel(const float* __restrict__ h, float* __restrict__ ws) {
  const int tid = threadIdx.x;
  const float* Yf = &ws[OFF_YB];
  for (int idx = tid; idx < 2*512; idx += blockDim.x) {
    int e = idx / 512, p = idx % 512;
    const float* w5 = &ws[e ? OFF_W5E2 : OFF_W5E1];
    const float* yr = &Yf[(size_t)(e*512 + p) * 128];
    float acc = w5[128];
    for (int k = 0; k < 128; ++k) acc += yr[k] * w5[k];
    ws[OFF_S + idx] = acc;
  }
  __syncthreads();
  for (int idx = tid; idx < 2*BB*NN; idx += blockDim.x) {
    int e = idx / (BB*NN), b = (idx / NN) % BB, n = idx % NN;
    float acc = 0.f;
    for (int m = 0; m < NN; ++m) acc += ws[OFF_S + e*512 + (b*NN + n)*NN + m];
    ws[OFF_SROW + idx] = acc;
  }
  __syncthreads();
  // covar = [e1 (24), ip (8)] with e1[b,n,d] = srow1[b,n]*Zsum[b,d]
  for (int idx = tid; idx < BB*NN*32; idx += blockDim.x) {
    int b = idx / (NN*32), n = (idx / 32) % NN, d = idx % 32;
    float val;
    if (d < 24) val = ws[OFF_SROW + b*NN + n] * ws[OFF_ZSUM + b*24 + d];
    else        val = ws[OFF_IP + (b*NN+n)*8 + (d-24)];
    ws[OFF_COVAR + idx] = val;
  }
  __syncthreads();
  for (int idx = tid; idx < BB*32; idx += blockDim.x) {
    int b = idx / 32, d = idx % 32;
    float acc = 0.f;
    for (int n = 0; n < NN; ++n) acc += ws[OFF_COVAR + (b*NN+n)*32 + d];
    ws[OFF_CSUM + idx] = acc;
  }
  // Decoder features: [gram, e2 (24), h (32), cent_h (32), pad 3]
  for (int r = tid; r < BB*NN; r += blockDim.x) {
    int b = r / NN, n = r % NN;
    float* row = &ws[OFF_FD + (size_t)r * 92];
    float g = 1.0f;
    for (int d = 0; d < 32; ++d) { float c = ws[OFF_COVAR + r*32 + d]; g += c*c; }
    row[0] = g;
    float s2 = ws[OFF_SROW + BB*NN + b*NN + n];
    for (int d = 0; d < 24; ++d) row[1+d]  = s2 * ws[OFF_ZSUM + b*24 + d];
    for (int d = 0; d < 32; ++d) row[25+d] = h[(b*NN+n)*HH + d];
    for (int d = 0; d < 32; ++d) row[57+d] = ws[OFF_CENTH + (b*NN+n)*HH + d];
    row[89] = 0.f; row[90] = 0.f; row[91] = 0.f;
  }
}

// ---------------------------------------------------------------------------
// Final: out[b,n,d] = (sum_o dec_out[b,n,o]) * (sum_m covar[b,m,d])
// ---------------------------------------------------------------------------
__global__ void __launch_bounds__(256)
final_kernel(float* __restrict__ out, float* __restrict__ ws) {
  __shared__ float s_sdec[BB*NN];
  const int tid = threadIdx.x;
  const float* Yf = &ws[OFF_YDB];
  const float* w5 = &ws[OFF_W5D];
  for (int r = tid; r < BB*NN; r += blockDim.x) {
    float acc = w5[128];
    const float* yr = &Yf[(size_t)r * 128];
    for (int k = 0; k < 128; ++k) acc += yr[k] * w5[k];
    s_sdec[r] = acc;
  }
  __syncthreads();
  for (int idx = tid; idx < BB*NN*32; idx += blockDim.x) {
    int b = idx / (NN*32), n = (idx / 32) % NN, d = idx % 32;
    out[idx] = s_sdec[b*NN+n] * ws[OFF_CSUM + b*32 + d];
  }
}

extern "C" void kernel_launch(void* const* d_in, const int* in_sizes, int n_in,
                              void* d_out, int out_size, void* d_ws, size_t ws_size,
                              hipStream_t stream) {
  (void)in_sizes; (void)n_in; (void)out_size; (void)ws_size;
  const float* z     = (const float*)d_in[0];
  const float* h     = (const float*)d_in[1];
  const float* masks = (const float*)d_in[3];   // d_in[2] = edges (unused)
  const float *e1W[5], *e1b[5], *e2W[5], *e2b[5], *dW[5], *db[5];
  for (int i = 0; i < 5; ++i) {
    e1W[i] = (const float*)d_in[4  + 2*i]; e1b[i] = (const float*)d_in[5  + 2*i];
    e2W[i] = (const float*)d_in[14 + 2*i]; e2b[i] = (const float*)d_in[15 + 2*i];
    dW[i]  = (const float*)d_in[24 + 2*i]; db[i]  = (const float*)d_in[25 + 2*i];
  }
  float* ws  = (float*)d_ws;
  float* out = (float*)d_out;

  prep_kernel<<<1, 256, 0, stream>>>(z, h, masks, e1W[0], e2W[0], dW[0],
                                     e1W[4], e1b[4], e2W[4], e2b[4], dW[4], db[4], ws);

  float* F    = ws + OFF_F;
  float* W1e1 = ws + OFF_W1E1;
  float* W1e2 = ws + OFF_W1E2;
  float* W1d  = ws + OFF_W1D;
  float* YA   = ws + OFF_YA;
  float* YB   = ws + OFF_YB;

  dim3 gEnc(32, 8, 2);  // 512 rows / 16, 128 cols / 16, enc1|enc2
  gemm_relu_wmma<<<gEnc, 32, 0, stream>>>(F, F, W1e1, W1e2, e1b[0], e2b[0],
                                          YA, YA + 512*128, 132, 132, 1);
  gemm_relu_wmma<<<gEnc, 32, 0, stream>>>(YA, YA + 512*128, e1W[1], e2W[1], e1b[1], e2b[1],
                                          YB, YB + 512*128, 128, 128, 1);
  gemm_relu_wmma<<<gEnc, 32, 0, stream>>>(YB, YB + 512*128, e1W[2], e2W[2], e1b[2], e2b[2],
                                          YA, YA + 512*128, 128, 128, 1);
  gemm_relu_wmma<<<gEnc, 32, 0, stream>>>(YA, YA + 512*128, e1W[3], e2W[3], e1b[3], e2b[3],
                                          YB, YB + 512*128, 128, 128, 1);

  combine_kernel<<<1, 256, 0, stream>>>(h, ws);

  float* Fd  = ws + OFF_FD;
  float* YDA = ws + OFF_YDA;
  float* YDB = ws + OFF_YDB;
  dim3 gDec(2, 8, 1);   // 32 rows / 16, 128 cols / 16
  gemm_relu_wmma<<<gDec, 32, 0, stream>>>(Fd, Fd, W1d, W1d, db[0], db[0],
                                          YDA, YDA, 92, 92, 1);
  gemm_relu_wmma<<<gDec, 32, 0, stream>>>(YDA, YDA, dW[1], dW[1], db[1], db[1],
                                          YDB, YDB, 128, 128, 1);
  gemm_relu_wmma<<<gDec, 32, 0, stream>>>(YDB, YDB, dW[2], dW[2], db[2], db[2],
                                          YDA, YDA, 128, 128, 1);
  gemm_relu_wmma<<<gDec, 32, 0, stream>>>(YDA, YDA, dW[3], dW[3], db[3], db[3],
                                          YDB, YDB, 128, 128, 1);

  final_kernel<<<1, 256, 0, stream>>>(out, ws);
}